// PrimitiveCNO_71743133713009
// MI455X (gfx1250) — compile-verified
//
#include <hip/hip_runtime.h>
#include <hip/hip_bf16.h>

#define B_  32
#define S_  4096
#define C_  64
#define H_  128
#define P_  8
#define KW  5

typedef __attribute__((ext_vector_type(2))) float v2f;
typedef __attribute__((ext_vector_type(8))) float v8f;

__device__ __forceinline__ v8f wmma_f32_k4(v2f a, v2f b, v8f c) {
  // D(16x16,f32) += A(16x4,f32) * B(4x16,f32); exact f32 path on CDNA5
  return __builtin_amdgcn_wmma_f32_16x16x4_f32(false, a, false, b, (short)0, c,
                                               false, false);
}

__device__ __forceinline__ float gelu_tanh(float x) {
  // JAX default gelu (approximate=True), branchless:
  //   0.5x(1+tanh(t)) == x * sigmoid(2t),  2t = 1.5957691216*(x + 0.044715 x^3)
  // __expf -> v_exp_f32 (saturates cleanly), rcp -> v_rcp_f32 (no branches)
  float x3 = x * x * x;
  float y  = 1.5957691216057308f * (x + 0.044715f * x3);
  return x * __builtin_amdgcn_rcpf(1.0f + __expf(-y));
}

// ---------------- gating: mean over S, logits, top-2, softmax ----------------
__global__ __launch_bounds__(256)
void moe_gate_kernel(const float* __restrict__ u,
                     const float* __restrict__ Wr,
                     const float* __restrict__ br,
                     int* __restrict__ gate_out) {
  const int b   = blockIdx.x;
  const int tid = threadIdx.x;
  const int c   = tid & 63;
  const int g   = tid >> 6;

  __shared__ float part[4][64];
  __shared__ float meanv[64];
  __shared__ float logits[P_];

  float s = 0.f;
  const float* ub = u + (size_t)b * S_ * C_;
  for (int srow = g; srow < S_; srow += 4) s += ub[srow * C_ + c];
  part[g][c] = s;
  __syncthreads();

  if (tid < 64)
    meanv[c] = (part[0][c] + part[1][c] + part[2][c] + part[3][c]) * (1.0f / S_);
  __syncthreads();

  if (tid < P_) {
    float l = br[tid];
    for (int cc = 0; cc < C_; ++cc) l += meanv[cc] * Wr[cc * P_ + tid];
    logits[tid] = l;
  }
  __syncthreads();

  if (tid == 0) {
    int i0 = 0; float v0 = logits[0];
    for (int p = 1; p < P_; ++p)
      if (logits[p] > v0) { v0 = logits[p]; i0 = p; }   // strict > keeps first idx on ties
    int i1 = -1; float v1 = -3.0e38f;
    for (int p = 0; p < P_; ++p)
      if (p != i0 && logits[p] > v1) { v1 = logits[p]; i1 = p; }
    float e1 = __expf(v1 - v0);          // v1 <= v0, safe
    float w0 = 1.0f / (1.0f + e1);
    float w1 = e1 * w0;
    int* o = gate_out + b * 4;
    o[0] = i0; o[1] = i1;
    o[2] = __float_as_int(w0); o[3] = __float_as_int(w1);
  }
}

// ---------------- main MoE conv kernel ----------------
// Grid: (S/64, B). Block: 256 threads = 8 wave32.
// LDS layout (floats) — u/h tiles FIRST so all A-fragment ds_loads get
// 16-bit immediate offsets (max h byte addr = 65,068 < 65,536):
//   u_lds [0, 5712)          84 rows x stride 68 (rows s0-10 .. s0+73, zero-padded)
//   h_lds [5712, 16272)      80 rows x stride 132 (h rows s0-8 .. s0+71)
//   wbuf  [16272, 67472)     staged weights (W1 then W2, pair-interleaved)
#define U_OFF    0
#define H_OFF    5712
#define WBUF_OFF 16272
#define LDS_F    67472

__global__ __launch_bounds__(256)
void moe_conv_kernel(const float* __restrict__ u,
                     const float* __restrict__ W1,
                     const float* __restrict__ b1,
                     const float* __restrict__ W2,
                     const float* __restrict__ b2,
                     const int*   __restrict__ gate,
                     float* __restrict__ out) {
  extern __shared__ float smem[];
  float* u_lds = smem + U_OFF;
  float* h_lds = smem + H_OFF;
  float* wbuf  = smem + WBUF_OFF;

  const int s0   = blockIdx.x * 64;
  const int b    = blockIdx.y;
  const int tid  = threadIdx.x;
  const int lane = tid & 31;
  const int wv   = tid >> 5;
  const int n    = lane & 15;   // column within 16x16 tile
  const int hi   = lane >> 4;   // half-wave select (K offset / row offset)

  // ---- load u halo tile into LDS (zero-padded SAME boundary) ----
  {
    const float* ub = u + (size_t)b * S_ * C_;
    for (int idx = tid; idx < 84 * 64; idx += 256) {
      int r = idx >> 6, c = idx & 63;
      int gs = s0 - 10 + r;
      float v = (gs >= 0 && gs < S_) ? ub[gs * C_ + c] : 0.f;
      u_lds[r * 68 + c] = v;
    }
  }

  // ---- per-batch routing (tiny, L2-resident) ----
  const int* gp = gate + b * 4;
  int   eidx[2] = { gp[0], gp[1] };
  float ew[2]   = { __int_as_float(gp[2]), __int_as_float(gp[3]) };
  __syncthreads();

  // conv2 / output tile assignment: wave -> (ct, two row tiles)
  const int ct  = wv & 3, c0 = ct * 16;
  const int rtb = wv >> 2;                // row tiles rtb and rtb+2

  // persistent output accumulator, init from u_state (= u here)
  float vout[2][8];
  #pragma unroll
  for (int j = 0; j < 2; ++j) {
    int rt = rtb + 2 * j;
    #pragma unroll
    for (int g = 0; g < 8; ++g) {
      int m = g + 8 * hi;
      vout[j][g] = u_lds[(rt * 16 + m + 10) * 68 + c0 + n];
    }
  }

  const int ht = wv, h0 = ht * 16;        // conv1 column tile per wave

  // per-lane A-fragment base pointers (row n, K sub-offset 2*hi) — the
  // remaining index terms are compile-time / loop constants -> DS immediates
  const float* aru = u_lds + n * 68 + 2 * hi;        // + (rt*16+k)*68 + cc
  const float* arh = h_lds + (n + 6) * 132 + 2 * hi; // + (rt*16+k)*132 + hh

  #pragma unroll 1
  for (int e = 0; e < 2; ++e) {
    const int   p  = eidx[e];
    const float we = ew[e];

    // ---- stage W1[p] -> LDS, pair-interleaved: (k*32 + c/2)*288 + h*2 + (c&1)
    {
      const float* w1p = W1 + (size_t)p * KW * C_ * H_;
      for (int idx = tid; idx < KW * C_ * H_; idx += 256) {
        int k = idx >> 13, rem = idx & 8191;
        int c = rem >> 7,  h   = rem & 127;
        wbuf[(k * 32 + (c >> 1)) * 288 + h * 2 + (c & 1)] = w1p[idx];
      }
    }
    __syncthreads();

    // ---- conv1 (5 shifted GEMMs) + bias + gelu -> h_lds ----
    {
      const float bb = b1[p * H_ + h0 + n];
      v8f z = {};
      v8f acc[5];
      #pragma unroll
      for (int rt = 0; rt < 5; ++rt) acc[rt] = z;

      #pragma unroll 1
      for (int k = 0; k < KW; ++k) {
        const float* wk = wbuf + ((k * 32 + hi) * 288) + (h0 + n) * 2;
        const float* ak = aru + k * 68;
        #pragma unroll 4
        for (int cc = 0; cc < C_; cc += 4) {
          v2f bfrag = *(const v2f*)(wk + (cc >> 1) * 288);   // rows cc+2*hi, +1
          #pragma unroll
          for (int rt = 0; rt < 5; ++rt) {                   // reuse B across 5 tiles
            v2f afrag = *(const v2f*)(ak + rt * (16 * 68) + cc);
            acc[rt] = wmma_f32_k4(afrag, bfrag, acc[rt]);
          }
        }
      }
      #pragma unroll
      for (int rt = 0; rt < 5; ++rt) {
        #pragma unroll
        for (int g = 0; g < 8; ++g) {
          int m = g + 8 * hi;
          int hrow = s0 - 8 + rt * 16 + m;                   // global h row
          float hv = (hrow >= 0 && hrow < S_) ? gelu_tanh(acc[rt][g] + bb) : 0.f;
          h_lds[(rt * 16 + m) * 132 + h0 + n] = hv;          // zero outside [0,S)
        }
      }
    }
    __syncthreads();

    // ---- stage W2[p] -> LDS: (k*64 + h/2)*160 + c*2 + (h&1) ----
    {
      const float* w2p = W2 + (size_t)p * KW * H_ * C_;
      for (int idx = tid; idx < KW * H_ * C_; idx += 256) {
        int k = idx >> 13, rem = idx & 8191;
        int h = rem >> 6,  c   = rem & 63;
        wbuf[(k * 64 + (h >> 1)) * 160 + c * 2 + (h & 1)] = w2p[idx];
      }
    }
    __syncthreads();

    // ---- conv2 (5 shifted GEMMs) + bias, weighted accumulate ----
    {
      const float bb2 = b2[p * C_ + c0 + n];
      v8f z = {};
      v8f acc2[2] = { z, z };

      #pragma unroll 1
      for (int k = 0; k < KW; ++k) {
        const float* wk = wbuf + ((k * 64 + hi) * 160) + (c0 + n) * 2;
        const float* ak = arh + (k + rtb * 16) * 132;
        #pragma unroll 4
        for (int hh = 0; hh < H_; hh += 4) {
          v2f bfrag = *(const v2f*)(wk + (hh >> 1) * 160);
          #pragma unroll
          for (int j = 0; j < 2; ++j) {
            v2f afrag = *(const v2f*)(ak + j * (32 * 132) + hh);
            acc2[j] = wmma_f32_k4(afrag, bfrag, acc2[j]);
          }
        }
      }
      #pragma unroll
      for (int j = 0; j < 2; ++j)
        #pragma unroll
        for (int g = 0; g < 8; ++g)
          vout[j][g] += we * (acc2[j][g] + bb2);
    }
    __syncthreads();   // wbuf/h_lds reuse safety for next expert
  }

  // ---- store output (always in range: tile fully inside [0,S)) ----
  #pragma unroll
  for (int j = 0; j < 2; ++j) {
    int rt = rtb + 2 * j;
    #pragma unroll
    for (int g = 0; g < 8; ++g) {
      int m = g + 8 * hi;
      out[((size_t)b * S_ + s0 + rt * 16 + m) * C_ + c0 + n] = vout[j][g];
    }
  }
}

// ---------------- launch ----------------
extern "C" void kernel_launch(void* const* d_in, const int* in_sizes, int n_in,
                              void* d_out, int out_size, void* d_ws, size_t ws_size,
                              hipStream_t stream) {
  const float* u  = (const float*)d_in[0];
  const float* W1 = (const float*)d_in[1];
  const float* b1 = (const float*)d_in[2];
  const float* W2 = (const float*)d_in[3];
  const float* b2 = (const float*)d_in[4];
  const float* Wr = (const float*)d_in[5];
  const float* br = (const float*)d_in[6];
  float* out = (float*)d_out;
  int*   gate = (int*)d_ws;   // 32 * 4 ints

  moe_gate_kernel<<<dim3(B_), dim3(256), 0, stream>>>(u, Wr, br, gate);

  size_t lds_bytes = (size_t)LDS_F * sizeof(float);   // ~264 KB, <= 320 KB/WGP
  moe_conv_kernel<<<dim3(S_ / 64, B_), dim3(256), lds_bytes, stream>>>(
      u, W1, b1, W2, b2, gate, out);
}